// SIGReg_6184752906274
// MI455X (gfx1250) — compile-verified
//
#include <hip/hip_runtime.h>
#include <math.h>

// Problem constants (match reference)
constexpr int MM = 1024;   // NUM_SLICES
constexpr int PP = 17;     // NUM_POINTS
constexpr int BN = 8192;   // B
constexpr int DD = 2048;   // D
constexpr float EPS = 1e-8f;
constexpr float T0 = 0.1f;
constexpr float DT = 1.9f / 16.0f;   // linspace(0.1, 2.0, 17) spacing

typedef __attribute__((ext_vector_type(2))) float v2f;
typedef __attribute__((ext_vector_type(8))) float v8f;

// ---------------------------------------------------------------------------
// Block-wide sum for 256-thread blocks (8 wave32). Returns total to all lanes.
// ---------------------------------------------------------------------------
__device__ __forceinline__ float block_sum256(float v, float* wred) {
#pragma unroll
    for (int off = 16; off > 0; off >>= 1) v += __shfl_xor(v, off, 32);
    const int lane = threadIdx.x & 31;
    const int w    = threadIdx.x >> 5;
    if (lane == 0) wred[w] = v;
    __syncthreads();
    float s = 0.f;
#pragma unroll
    for (int i = 0; i < 8; ++i) s += wred[i];
    __syncthreads();
    return s;
}

// ---------------------------------------------------------------------------
// Kernel 1: normalize direction rows: dnorm[m] = dirs[m] / (||dirs[m]|| + EPS)
// grid = MM blocks, 256 threads; each thread handles 8 elements (2048/256).
// ---------------------------------------------------------------------------
__global__ __launch_bounds__(256) void norm_dirs_k(const float* __restrict__ dirs,
                                                   float* __restrict__ dnorm) {
    __shared__ float wred[8];
    const int m   = blockIdx.x;
    const int tid = threadIdx.x;
    const float* r = dirs + (size_t)m * DD;

    const float4 va = *(const float4*)(r + tid * 4);
    const float4 vb = *(const float4*)(r + 1024 + tid * 4);
    float ss = va.x * va.x + va.y * va.y + va.z * va.z + va.w * va.w +
               vb.x * vb.x + vb.y * vb.y + vb.z * vb.z + vb.w * vb.w;
    const float tot = block_sum256(ss, wred);
    const float scale = 1.0f / (sqrtf(tot) + EPS);

    float* o = dnorm + (size_t)m * DD;
    float4 oa = {va.x * scale, va.y * scale, va.z * scale, va.w * scale};
    float4 ob = {vb.x * scale, vb.y * scale, vb.z * scale, vb.w * scale};
    *(float4*)(o + tid * 4) = oa;
    *(float4*)(o + 1024 + tid * 4) = ob;
}

// ---------------------------------------------------------------------------
// Kernel 2: proj[m][b] = sum_d dnorm[m][d] * z[b][d], via V_WMMA_F32_16X16X4_F32
// Block: 256 threads = 8 waves. Block tile 64(M) x 128(B), K-panel = 32.
// Wave (wm 0..3, wn 0..1) computes rows [wm*16,16) x cols [wn*64,64)
// as 4 16x16 WMMA accumulators sharing one A fragment.
// ---------------------------------------------------------------------------
constexpr int MT = 64;
constexpr int NT = 128;
constexpr int KT = 32;

__global__ __launch_bounds__(256) void gemm_proj_k(const float* __restrict__ dn,
                                                   const float* __restrict__ z,
                                                   float* __restrict__ proj) {
    __shared__ float As[MT][KT + 1];   // [m][k], padded stride 33
    __shared__ float Bs[NT][KT + 1];   // [b][k] = z tile row-major in k

    const int tid  = threadIdx.x;
    const int m0   = blockIdx.y * MT;
    const int n0   = blockIdx.x * NT;
    const int lane = tid & 31;
    const int w    = tid >> 5;
    const int wm   = w >> 1;       // 0..3
    const int wn   = w & 1;        // 0..1
    const int half = lane >> 4;    // 0 or 1
    const int lr   = lane & 15;

    v8f acc0 = {}, acc1 = {}, acc2 = {}, acc3 = {};

    for (int k0 = 0; k0 < DD; k0 += KT) {
        // Stage A tile: 64x32 = 512 float4, 2 per thread (coalesced on k)
#pragma unroll
        for (int i = 0; i < 2; ++i) {
            const int idx = tid + i * 256;
            const int r  = idx >> 3;
            const int c4 = (idx & 7) << 2;
            const float4 v = *(const float4*)(dn + (size_t)(m0 + r) * DD + k0 + c4);
            As[r][c4 + 0] = v.x; As[r][c4 + 1] = v.y;
            As[r][c4 + 2] = v.z; As[r][c4 + 3] = v.w;
        }
        // Stage B tile: 128x32 = 1024 float4, 4 per thread
#pragma unroll
        for (int i = 0; i < 4; ++i) {
            const int idx = tid + i * 256;
            const int r  = idx >> 3;
            const int c4 = (idx & 7) << 2;
            const float4 v = *(const float4*)(z + (size_t)(n0 + r) * DD + k0 + c4);
            Bs[r][c4 + 0] = v.x; Bs[r][c4 + 1] = v.y;
            Bs[r][c4 + 2] = v.z; Bs[r][c4 + 3] = v.w;
        }
        __syncthreads();

#pragma unroll
        for (int kk = 0; kk < KT; kk += 4) {
            const int ka = kk + 2 * half;   // lanes 0-15: K=kk,kk+1; lanes 16-31: K=kk+2,kk+3
            // A fragment (16x4): row = wm*16 + lr
            v2f a;
            a.x = As[wm * 16 + lr][ka];
            a.y = As[wm * 16 + lr][ka + 1];
            // B fragments (4x16): B[k][n] = z[n][k] -> Bs[n][k]
            const int nb = wn * 64 + lr;
            v2f b0, b1, b2, b3;
            b0.x = Bs[nb +  0][ka]; b0.y = Bs[nb +  0][ka + 1];
            b1.x = Bs[nb + 16][ka]; b1.y = Bs[nb + 16][ka + 1];
            b2.x = Bs[nb + 32][ka]; b2.y = Bs[nb + 32][ka + 1];
            b3.x = Bs[nb + 48][ka]; b3.y = Bs[nb + 48][ka + 1];

            acc0 = __builtin_amdgcn_wmma_f32_16x16x4_f32(false, a, false, b0, (short)0, acc0, false, false);
            acc1 = __builtin_amdgcn_wmma_f32_16x16x4_f32(false, a, false, b1, (short)0, acc1, false, false);
            acc2 = __builtin_amdgcn_wmma_f32_16x16x4_f32(false, a, false, b2, (short)0, acc2, false, false);
            acc3 = __builtin_amdgcn_wmma_f32_16x16x4_f32(false, a, false, b3, (short)0, acc3, false, false);
        }
        __syncthreads();
    }

    // Epilogue: D layout: VGPR v -> M = v + 8*half, N = lr
#pragma unroll
    for (int v = 0; v < 8; ++v) {
        const int mrow = m0 + wm * 16 + v + 8 * half;
        const size_t base = (size_t)mrow * BN + n0 + wn * 64 + lr;
        proj[base +  0] = acc0[v];
        proj[base + 16] = acc1[v];
        proj[base + 32] = acc2[v];
        proj[base + 48] = acc3[v];
    }
}

// ---------------------------------------------------------------------------
// Kernel 3: per slice m: mean/std (ddof=1) over B, then ECF means for 17
// equally-spaced t using a sincos rotation recurrence (2 sincos per element).
// grid = MM blocks, 256 threads. Row (32 KB) cached in LDS.
// ecf layout: ecf[m*34 + p] = real mean, ecf[m*34 + 17 + p] = imag mean.
// ---------------------------------------------------------------------------
__global__ __launch_bounds__(256) void slice_ecf_k(const float* __restrict__ proj,
                                                   float* __restrict__ ecf) {
    __shared__ __align__(16) float row[BN];
    __shared__ float wred[8];

    const int m   = blockIdx.x;
    const int tid = threadIdx.x;
    const float* pr = proj + (size_t)m * BN;

    // Load row into LDS, accumulating sum and sumsq on the fly.
    float lsum = 0.f, lsq = 0.f;
#pragma unroll
    for (int i = 0; i < 8; ++i) {
        const int idx4 = tid + i * 256;           // float4 index, 2048 total
        const float4 v = *(const float4*)(pr + idx4 * 4);
        *(float4*)(row + idx4 * 4) = v;
        lsum += v.x + v.y + v.z + v.w;
        lsq  += v.x * v.x + v.y * v.y + v.z * v.z + v.w * v.w;
    }
    __syncthreads();

    const float sum = block_sum256(lsum, wred);
    const float sq  = block_sum256(lsq, wred);
    const float mean = sum / (float)BN;
    float var = (sq - sum * mean) / (float)(BN - 1);
    var = fmaxf(var, 0.0f);
    const float istd = 1.0f / (sqrtf(var) + EPS);

    // ECF accumulation: cos/sin(t_p * s) with t_p = T0 + p*DT.
    float cr[PP], ci[PP];
#pragma unroll
    for (int p = 0; p < PP; ++p) { cr[p] = 0.f; ci[p] = 0.f; }

    for (int b = tid; b < BN; b += 256) {
        const float s = (row[b] - mean) * istd;
        float sb, cb, sd, cd;
        __sincosf(T0 * s, &sb, &cb);   // angle = t_0 * s
        __sincosf(DT * s, &sd, &cd);   // rotation step = dt * s
#pragma unroll
        for (int p = 0; p < PP; ++p) {
            cr[p] += cb;
            ci[p] += sb;
            const float c2 = cb * cd - sb * sd;
            const float s2 = sb * cd + cb * sd;
            cb = c2; sb = s2;
        }
    }

    // Reduce 34 accumulators across the block; thread 0 writes the means.
    const float invB = 1.0f / (float)BN;
#pragma unroll
    for (int p = 0; p < PP; ++p) {
        const float tr = block_sum256(cr[p], wred);
        const float ti = block_sum256(ci[p], wred);
        if (tid == 0) {
            ecf[(size_t)m * (2 * PP) + p]      = tr * invB;
            ecf[(size_t)m * (2 * PP) + PP + p] = ti * invB;
        }
    }
}

// ---------------------------------------------------------------------------
// Kernel 4: stat = mean over (m,p) of (ecf_r - exp(-t^2/2))^2 + ecf_i^2
// ---------------------------------------------------------------------------
__global__ __launch_bounds__(256) void finalize_k(const float* __restrict__ ecf,
                                                  float* __restrict__ out) {
    __shared__ float wred[8];
    float acc = 0.f;
    for (int i = threadIdx.x; i < MM * PP; i += 256) {
        const int mi = i / PP;
        const int p  = i - mi * PP;
        const float t = T0 + (float)p * DT;
        const float g = expf(-0.5f * t * t);
        const float er = ecf[(size_t)mi * (2 * PP) + p];
        const float ei = ecf[(size_t)mi * (2 * PP) + PP + p];
        const float d = er - g;
        acc += d * d + ei * ei;
    }
    const float tot = block_sum256(acc, wred);
    if (threadIdx.x == 0) out[0] = tot / (float)(MM * PP);
}

// ---------------------------------------------------------------------------
extern "C" void kernel_launch(void* const* d_in, const int* in_sizes, int n_in,
                              void* d_out, int out_size, void* d_ws, size_t ws_size,
                              hipStream_t stream) {
    const float* z    = (const float*)d_in[0];   // (B, D) fp32
    const float* dirs = (const float*)d_in[1];   // (M, D) fp32
    float* out = (float*)d_out;

    float* dnorm = (float*)d_ws;                      // M*D      = 2,097,152 f
    float* proj  = dnorm + (size_t)MM * DD;           // M*B      = 8,388,608 f
    float* ecf   = proj + (size_t)MM * BN;            // M*2*P    = 34,816 f

    norm_dirs_k<<<MM, 256, 0, stream>>>(dirs, dnorm);
    gemm_proj_k<<<dim3(BN / NT, MM / MT), 256, 0, stream>>>(dnorm, z, proj);
    slice_ecf_k<<<MM, 256, 0, stream>>>(proj, ecf);
    finalize_k<<<1, 256, 0, stream>>>(ecf, out);
}